// ImbalancedGAT_43456479101293
// MI455X (gfx1250) — compile-verified
//
#include <hip/hip_runtime.h>
#include <hip/hip_bf16.h>
#include <float.h>
#include <math.h>

typedef __attribute__((ext_vector_type(2))) float v2f;
typedef __attribute__((ext_vector_type(8))) float v8f;

#define GAT_N     50000
#define GAT_INCH  128
#define GAT_HID   64
#define GAT_HEADS 2
#define GAT_D1    128   /* HEADS*HID */
#define GAT_OUT   2

// ---------------------------------------------------------------- utilities
__global__ void k_fill(float* __restrict__ p, long n, float v) {
  long i = (long)blockIdx.x * blockDim.x + threadIdx.x;
  if (i < n) p[i] = v;
}

__device__ inline void atomicMaxF(float* addr, float v) {
  // monotonic under IEEE bit patterns split by sign
  if (v >= 0.0f) atomicMax((int*)addr, __float_as_int(v));
  else           atomicMin((unsigned int*)addr, __float_as_uint(v));
}

// ------------------------------------------------- WMMA f32 GEMM  Y = X * W
// X: [M,128] row-major, W: [128,128] row-major, Y: [M,128].
// One wave -> one 16x16 tile of Y, K stepped by 4 via V_WMMA_F32_16X16X4_F32.
__global__ void k_gemm_wmma_f32(const float* __restrict__ X,
                                const float* __restrict__ W,
                                float* __restrict__ Y, int M) {
  const int wave = threadIdx.x >> 5;
  const int lane = threadIdx.x & 31;
  const int tile = blockIdx.x * (blockDim.x >> 5) + wave;
  const int mtiles = M >> 4;              // M multiple of 16
  if (tile >= mtiles * 8) return;
  const int mt = tile >> 3;               // 8 col tiles (128/16)
  const int nt = tile & 7;
  const int row0 = mt << 4;
  const int col0 = nt << 4;
  const int hl = lane & 15;               // 0..15
  const int hi = lane >> 4;               // 0 (K,K+1) / 1 (K+2,K+3); rows M vs M+8 for C/D

  v8f c = {};
  const float* xrow = X + (size_t)(row0 + hl) * GAT_INCH;   // A: M = hl
  #pragma unroll 4
  for (int k = 0; k < GAT_INCH; k += 4) {
    v2f a, b;
    // A 16x4: VGPR0 = K (lanes 0-15) / K+2 (lanes 16-31); VGPR1 = K+1 / K+3
    a.x = xrow[k + 2 * hi];
    a.y = xrow[k + 2 * hi + 1];
    // B 4x16: VGPR0 = row K (lanes 0-15) / row K+2 (lanes 16-31); VGPR1 = K+1 / K+3
    b.x = W[(size_t)(k + 2 * hi) * GAT_D1 + col0 + hl];
    b.y = W[(size_t)(k + 2 * hi + 1) * GAT_D1 + col0 + hl];
    c = __builtin_amdgcn_wmma_f32_16x16x4_f32(false, a, false, b, (short)0, c,
                                              false, false);
  }
  // C/D 16x16: VGPR r -> M = row0 + r + 8*hi, N = col0 + hl
  #pragma unroll
  for (int r = 0; r < 8; ++r)
    Y[(size_t)(row0 + 8 * hi + r) * GAT_D1 + col0 + hl] = c[r];
}

// ------------------------------------------- per-node attention projections
__global__ void k_node_alpha(const float* __restrict__ xw,
                             const float* __restrict__ a_src,
                             const float* __restrict__ a_dst,
                             float* __restrict__ as_, float* __restrict__ ad_,
                             int N, int H, int C) {
  int n = blockIdx.x * blockDim.x + threadIdx.x;
  if (n >= N) return;
  for (int h = 0; h < H; ++h) {
    const float* row = xw + (size_t)n * H * C + (size_t)h * C;
    float s = 0.f, d = 0.f;
    for (int cc = 0; cc < C; ++cc) {
      s += row[cc] * a_src[h * C + cc];
      d += row[cc] * a_dst[h * C + cc];
    }
    as_[n * H + h] = s;
    ad_[n * H + h] = d;
  }
}

// --------------------------------- edge pass 1: leaky-relu logit + seg max
__global__ void k_edge_logit_max(const int* __restrict__ ei, int E, int N,
                                 const float* __restrict__ as_,
                                 const float* __restrict__ ad_,
                                 float* __restrict__ elog,
                                 float* __restrict__ smax, int H) {
  int e = blockIdx.x * blockDim.x + threadIdx.x;
  int ET = E + N;
  if (e >= ET) return;
  int s = (e < E) ? ei[e] : (e - E);
  int d = (e < E) ? ei[E + e] : (e - E);
  for (int h = 0; h < H; ++h) {
    float v = as_[s * H + h] + ad_[d * H + h];
    v = (v > 0.f) ? v : 0.2f * v;               // LeakyReLU(0.2)
    elog[(size_t)e * H + h] = v;
    atomicMaxF(&smax[d * H + h], v);
  }
}

// ----------------------------------- edge pass 2: exp(e - max) + seg denom
__global__ void k_edge_exp(const int* __restrict__ ei, int E, int N,
                           float* __restrict__ elog,
                           const float* __restrict__ smax,
                           float* __restrict__ den, int H) {
  int e = blockIdx.x * blockDim.x + threadIdx.x;
  int ET = E + N;
  if (e >= ET) return;
  int d = (e < E) ? ei[E + e] : (e - E);
  for (int h = 0; h < H; ++h) {
    float ex = expf(elog[(size_t)e * H + h] - smax[d * H + h]);
    elog[(size_t)e * H + h] = ex;
    atomicAdd(&den[d * H + h], ex);
  }
}

// ----------------- edge pass 3 (layer 1): warp/edge weighted scatter, C=128
__global__ void k_edge_agg128(const int* __restrict__ ei, int E, int N,
                              const float* __restrict__ elog,
                              const float* __restrict__ den,
                              const float* __restrict__ xw,
                              float* __restrict__ agg) {
  long idx = (long)blockIdx.x * blockDim.x + threadIdx.x;
  int e = (int)(idx >> 5);
  int lane = (int)(idx & 31);
  int ET = E + N;
  if (e >= ET) return;
  int s = (e < E) ? ei[e] : (e - E);
  int d = (e < E) ? ei[E + e] : (e - E);
  float a0 = elog[(size_t)e * 2 + 0] / (den[d * 2 + 0] + 1e-16f);
  float a1 = elog[(size_t)e * 2 + 1] / (den[d * 2 + 1] + 1e-16f);
  const float* xr = xw + (size_t)s * GAT_D1;
  float* ar = agg + (size_t)d * GAT_D1;
  __builtin_prefetch(xr, 0, 0);                 // global_prefetch_b8
  #pragma unroll
  for (int i = 0; i < 4; ++i) {
    int c = lane + i * 32;                      // head = c/64
    float al = (c < GAT_HID) ? a0 : a1;
    atomicAdd(&ar[c], xr[c] * al);
  }
}

// ---- finalize layer1: +bias, ELU, project to layer2 (xw2, alpha_src/dst 2)
__global__ void k_finalize1(const float* __restrict__ agg,
                            const float* __restrict__ b1,
                            const float* __restrict__ W2,
                            const float* __restrict__ as2w,
                            const float* __restrict__ ad2w,
                            float* __restrict__ xw2,
                            float* __restrict__ as2, float* __restrict__ ad2,
                            int N) {
  int n = blockIdx.x * blockDim.x + threadIdx.x;
  if (n >= N) return;
  const float* row = agg + (size_t)n * GAT_D1;
  float o0 = 0.f, o1 = 0.f;
  for (int c = 0; c < GAT_D1; ++c) {
    float v = row[c] + b1[c];
    v = (v > 0.f) ? v : (expf(v) - 1.0f);       // ELU
    o0 += v * W2[c * 2 + 0];
    o1 += v * W2[c * 2 + 1];
  }
  xw2[n * 2 + 0] = o0;
  xw2[n * 2 + 1] = o1;
  as2[n] = o0 * as2w[0] + o1 * as2w[1];
  ad2[n] = o0 * ad2w[0] + o1 * ad2w[1];
}

// --------------------- edge pass 3 (layer 2): thread/edge, 2 channels, H=1
__global__ void k_edge_agg2(const int* __restrict__ ei, int E, int N,
                            const float* __restrict__ elog,
                            const float* __restrict__ den,
                            const float* __restrict__ xw2,
                            float* __restrict__ agg2) {
  int e = blockIdx.x * blockDim.x + threadIdx.x;
  int ET = E + N;
  if (e >= ET) return;
  int s = (e < E) ? ei[e] : (e - E);
  int d = (e < E) ? ei[E + e] : (e - E);
  float al = elog[e] / (den[d] + 1e-16f);
  atomicAdd(&agg2[d * 2 + 0], xw2[s * 2 + 0] * al);
  atomicAdd(&agg2[d * 2 + 1], xw2[s * 2 + 1] * al);
}

__global__ void k_finalize2(const float* __restrict__ agg2,
                            const float* __restrict__ b2,
                            float* __restrict__ out, int N) {
  int i = blockIdx.x * blockDim.x + threadIdx.x;
  if (i >= N * 2) return;
  out[i] = agg2[i] + b2[i & 1];
}

// --------------------------------------------------------------------------
extern "C" void kernel_launch(void* const* d_in, const int* in_sizes, int n_in,
                              void* d_out, int out_size, void* d_ws, size_t ws_size,
                              hipStream_t stream) {
  const float* x      = (const float*)d_in[0];
  const int*   ei     = (const int*)d_in[1];
  const float* W1     = (const float*)d_in[2];
  const float* a_src1 = (const float*)d_in[3];
  const float* a_dst1 = (const float*)d_in[4];
  const float* b1     = (const float*)d_in[5];
  const float* W2     = (const float*)d_in[6];
  const float* a_src2 = (const float*)d_in[7];
  const float* a_dst2 = (const float*)d_in[8];
  const float* b2     = (const float*)d_in[9];
  float* out = (float*)d_out;

  const int N  = GAT_N;
  const int E  = in_sizes[1] / 2;
  const int ET = E + N;

  // ---- workspace layout (floats)
  float* w = (float*)d_ws;
  size_t off = 0;
  float* xw1  = w + off; off += (size_t)N * GAT_D1;
  float* agg1 = w + off; off += (size_t)N * GAT_D1;
  float* as1  = w + off; off += (size_t)N * 2;
  float* ad1  = w + off; off += (size_t)N * 2;
  float* mx1  = w + off; off += (size_t)N * 2;
  float* dn1  = w + off; off += (size_t)N * 2;
  float* elog = w + off; off += (size_t)ET * 2;   // reused (size ET) for layer 2
  float* xw2  = w + off; off += (size_t)N * 2;
  float* as2  = w + off; off += (size_t)N;
  float* ad2  = w + off; off += (size_t)N;
  float* mx2  = w + off; off += (size_t)N;
  float* dn2  = w + off; off += (size_t)N;
  float* agg2 = w + off; off += (size_t)N * 2;
  (void)ws_size; (void)n_in; (void)out_size;

  const int T = 256;
  auto nb = [](long n, int t) { return (unsigned)((n + t - 1) / t); };

  // ---- init accumulators
  k_fill<<<nb((long)N * GAT_D1, T), T, 0, stream>>>(agg1, (long)N * GAT_D1, 0.f);
  k_fill<<<nb((long)N * 2, T), T, 0, stream>>>(mx1, (long)N * 2, -FLT_MAX);
  k_fill<<<nb((long)N * 2, T), T, 0, stream>>>(dn1, (long)N * 2, 0.f);
  k_fill<<<nb((long)N, T), T, 0, stream>>>(mx2, (long)N, -FLT_MAX);
  k_fill<<<nb((long)N, T), T, 0, stream>>>(dn2, (long)N, 0.f);
  k_fill<<<nb((long)N * 2, T), T, 0, stream>>>(agg2, (long)N * 2, 0.f);

  // ---- layer 1
  {
    int tiles = (N >> 4) * 8;                            // 25000 (N%16==0)
    int wpb = 128 >> 5;                                  // 4 waves/block
    k_gemm_wmma_f32<<<nb(tiles, wpb), 128, 0, stream>>>(x, W1, xw1, N);
  }
  k_node_alpha<<<nb(N, T), T, 0, stream>>>(xw1, a_src1, a_dst1, as1, ad1,
                                           N, GAT_HEADS, GAT_HID);
  k_edge_logit_max<<<nb(ET, T), T, 0, stream>>>(ei, E, N, as1, ad1, elog, mx1, 2);
  k_edge_exp<<<nb(ET, T), T, 0, stream>>>(ei, E, N, elog, mx1, dn1, 2);
  k_edge_agg128<<<nb((long)ET * 32, T), T, 0, stream>>>(ei, E, N, elog, dn1,
                                                        xw1, agg1);
  k_finalize1<<<nb(N, T), T, 0, stream>>>(agg1, b1, W2, a_src2, a_dst2,
                                          xw2, as2, ad2, N);

  // ---- layer 2 (1 head, 2 channels); reuse elog
  k_edge_logit_max<<<nb(ET, T), T, 0, stream>>>(ei, E, N, as2, ad2, elog, mx2, 1);
  k_edge_exp<<<nb(ET, T), T, 0, stream>>>(ei, E, N, elog, mx2, dn2, 1);
  k_edge_agg2<<<nb(ET, T), T, 0, stream>>>(ei, E, N, elog, dn2, xw2, agg2);
  k_finalize2<<<nb((long)N * 2, T), T, 0, stream>>>(agg2, b2, out, N);
}